// QuantizeEMAReset_60876866454144
// MI455X (gfx1250) — compile-verified
//
#include <hip/hip_runtime.h>
#include <hip/hip_bf16.h>
#include <math.h>

typedef __attribute__((ext_vector_type(2))) float v2f;
typedef __attribute__((ext_vector_type(4))) float v4f;
typedef __attribute__((ext_vector_type(8))) float v8f;
typedef __attribute__((ext_vector_type(4))) unsigned u32x4;
typedef __attribute__((ext_vector_type(8))) int i32x8;
typedef __attribute__((ext_vector_type(4))) int i32x4;

#define C_DIM 64
#define TOK_PER_BLOCK 128   // 8 waves * 16 tokens
#define KSTAGE 128          // codewords per LDS stage (32 KB), double-buffered
#define K_MAX 1024          // codebook rows (reference: 1024)

// ---------------------------------------------------------------------------
// Kernel 0: zero the histogram counts and the commit-loss accumulator
// ---------------------------------------------------------------------------
__global__ void vq_init_kernel(unsigned* __restrict__ counts,
                               float* __restrict__ commit_sum, int K) {
    int t = blockIdx.x * blockDim.x + threadIdx.x;
    if (t < K) counts[t] = 0u;
    if (t == 0) *commit_sum = 0.0f;
}

// ---------------------------------------------------------------------------
// Kernel 1: per-codeword squared norms cc[k] = sum_c codebook[k][c]^2
// ---------------------------------------------------------------------------
__global__ void vq_ccnorm_kernel(const float* __restrict__ cb,
                                 float* __restrict__ ccn, int K) {
    int k = blockIdx.x * blockDim.x + threadIdx.x;
    if (k >= K) return;
    const float* row = cb + (size_t)k * C_DIM;
    float s = 0.0f;
#pragma unroll
    for (int c = 0; c < C_DIM; c += 4) {
        v4f v = *(const v4f*)(row + c);
        s += v[0]*v[0] + v[1]*v[1] + v[2]*v[2] + v[3]*v[3];
    }
    ccn[k] = s;
}

// ---------------------------------------------------------------------------
// TDM: DMA one codebook stage (KSTAGE rows x 64 fp32, row stride 64) into LDS.
// D# per CDNA5 ISA ch.8: group0 = {count=1 | lds_addr | global_addr | type=2},
// group1 = {data_size=4B, tensor_dim0=64, tensor_dim1=128, tile 64x128,
//           tensor_dim0_stride=64, workgroup_mask=0}, groups 2/3 = 0 (2-D).
// 6-arg builtin form (clang-23 / therock-10.0 headers).
// ---------------------------------------------------------------------------
__device__ __forceinline__ void tdm_load_cb_stage(const float* gsrc,
                                                  unsigned lds_off) {
    unsigned long long ga = (unsigned long long)(size_t)gsrc;
    u32x4 g0;
    g0[0] = 1u;                                         // count=1 (valid), user
    g0[1] = lds_off;                                    // lds_addr (bytes)
    g0[2] = (unsigned)(ga & 0xFFFFFFFFu);               // global_addr[31:0]
    g0[3] = (unsigned)((ga >> 32) & 0x01FFFFFFu)        // global_addr[56:32]
            | (2u << 30);                               // type=2 ("image")
    i32x8 g1;
    g1[0] = (int)(2u << 16);        // wg_mask=0, data_size=2 (4 bytes)
    g1[1] = (int)(64u << 16);       // tensor_dim0 = 64      (bits 79:48 lo)
    g1[2] = (int)(128u << 16);      // tensor_dim1 = 128     (bits 111:80 lo)
    g1[3] = (int)(64u << 16);       // tile_dim0 = 64        (bits 127:112)
    g1[4] = (int)(KSTAGE);          // tile_dim1 = 128, tile_dim2 = 0
    g1[5] = (int)(C_DIM);           // tensor_dim0_stride = 64 (bits 191:160)
    g1[6] = 0;                      // stride hi / tensor_dim1_stride lo
    g1[7] = 0;
    i32x4 zz4 = {0, 0, 0, 0};
    i32x8 zz8 = {0, 0, 0, 0, 0, 0, 0, 0};
    __builtin_amdgcn_tensor_load_to_lds(g0, g1, zz4, zz4, zz8, 0);
}

// ---------------------------------------------------------------------------
// Kernel 2 (fused): fp32 WMMA distance + argmin + z_q gather + commit loss +
// histogram. Wave w of each 256-thread block owns tokens
// [blockIdx*128 + 16w, +16).
// A fragment (16x4 f32): lane L -> row  M = L%16, K-pair = 2*(L/16).
// B fragment (4x16 f32): lane L -> col  N = L%16, K-pair = 2*(L/16).
// C/D (16x16 f32, 8 VGPRs): lane L, reg r -> N = L%16, M = r + 8*(L/16).
// Codebook stages are TDM-DMA'd into double-buffered LDS by wave 0 and
// synchronized with s_wait_tensorcnt + workgroup barriers. Codeword norms are
// staged once into LDS so the per-tile argmin update never waits on global.
// ---------------------------------------------------------------------------
__global__ __launch_bounds__(256) void vq_main_kernel(
    const float* __restrict__ z,        // [N, 64]
    const float* __restrict__ cb,       // [K, 64]
    const float* __restrict__ ccn,      // [K]
    float* __restrict__ zq_out,         // [N, 64]
    float* __restrict__ commit_sum,
    unsigned* __restrict__ counts,
    int K) {
    __shared__ float lds_cb[2][KSTAGE * C_DIM];   // 2 x 32 KB
    __shared__ float ccn_sh[K_MAX];               // 4 KB
    __shared__ int   idx_sh[TOK_PER_BLOCK];
    __shared__ float red[256];

    const int tid  = threadIdx.x;
    const int wave = tid >> 5;
    const int lane = tid & 31;
    const int half = lane >> 4;   // 0 or 1
    const int l16  = lane & 15;
    const int tok0 = blockIdx.x * TOK_PER_BLOCK;
    const int m0   = tok0 + wave * 16;

    // Stage all codeword norms into LDS (covered by the first stage barrier).
    for (int i = tid; i < K; i += 256) ccn_sh[i] = ccn[i];

    // Preload the wave's A fragments (its 16 token rows, all 64 channels).
    const float* zrow = z + (size_t)(m0 + l16) * C_DIM;
    v2f afrag[16];
#pragma unroll
    for (int kk = 0; kk < 16; ++kk) {
        afrag[kk] = *(const v2f*)(zrow + 4 * kk + 2 * half);
    }

    float best[8];
    int   bidx[8];
#pragma unroll
    for (int r = 0; r < 8; ++r) { best[r] = 3.4e38f; bidx[r] = 0; }

    const unsigned lds_base = (unsigned)(size_t)(void*)&lds_cb[0][0];
    const int nstages = K / KSTAGE;

    if (wave == 0) tdm_load_cb_stage(cb, lds_base);          // prefetch stage 0

    for (int s = 0; s < nstages; ++s) {
        const int buf = s & 1;
        if (wave == 0) {
            if (s + 1 < nstages) {
                // Queue next stage into the other buffer, then wait until only
                // it is outstanding (TDM ops complete in order per wave).
                tdm_load_cb_stage(cb + (size_t)(s + 1) * KSTAGE * C_DIM,
                                  lds_base + (unsigned)((buf ^ 1) *
                                      KSTAGE * C_DIM * sizeof(float)));
                __builtin_amdgcn_s_wait_tensorcnt(1);
            } else {
                __builtin_amdgcn_s_wait_tensorcnt(0);
            }
        }
        __syncthreads();   // stage s (and ccn_sh on s==0) resident for all waves

#pragma unroll
        for (int t = 0; t < KSTAGE / 16; ++t) {     // 8 tiles of 16 codewords
            const int n0 = s * KSTAGE + t * 16;
            const float* brow = &lds_cb[buf][(t * 16 + l16) * C_DIM];

            v8f acc0 = {};
            v8f acc1 = {};
#pragma unroll
            for (int kk = 0; kk < 16; kk += 2) {    // C = 64 in K-steps of 4
                v2f b0 = *(const v2f*)(brow + 4 * kk + 2 * half);
                v2f b1 = *(const v2f*)(brow + 4 * (kk + 1) + 2 * half);
                acc0 = __builtin_amdgcn_wmma_f32_16x16x4_f32(
                    false, afrag[kk], false, b0, (short)0, acc0, false, false);
                acc1 = __builtin_amdgcn_wmma_f32_16x16x4_f32(
                    false, afrag[kk + 1], false, b1, (short)0, acc1, false, false);
            }

            const float ccl = ccn_sh[n0 + l16];     // ||c_n||^2 (LDS, no stall)
            const int   idx = n0 + l16;
#pragma unroll
            for (int r = 0; r < 8; ++r) {
                float d = ccl - 2.0f * (acc0[r] + acc1[r]);  // ||x||^2 dropped
                if (d < best[r]) { best[r] = d; bidx[r] = idx; }
            }
        }
        __syncthreads();   // all reads of buf done before it is re-DMA'd
    }

    // Cross-lane argmin within each 16-lane half (each half owns 8 token rows).
#pragma unroll
    for (int r = 0; r < 8; ++r) {
        float v = best[r];
        int   i = bidx[r];
#pragma unroll
        for (int off = 8; off >= 1; off >>= 1) {
            float ov = __shfl_xor(v, off, 32);
            int   oi = __shfl_xor(i, off, 32);
            if (ov < v || (ov == v && oi < i)) { v = ov; i = oi; }  // first-min
        }
        if (l16 == 0) idx_sh[wave * 16 + r + 8 * half] = i;
    }
    __syncthreads();

    // Cooperative tail: stream z_q, commit-loss partials, histogram.
    float local = 0.0f;
#pragma unroll
    for (int it = 0; it < (TOK_PER_BLOCK * 16) / 256; ++it) {   // 8 iters
        const int i    = it * 256 + tid;
        const int tok  = i >> 4;
        const int quad = i & 15;
        const int idx  = idx_sh[tok];
        v4f cvec = *(const v4f*)(cb + (size_t)idx * C_DIM + quad * 4);
        v4f xvec = *(const v4f*)(z  + (size_t)(tok0 + tok) * C_DIM + quad * 4);
        *(v4f*)(zq_out + (size_t)(tok0 + tok) * C_DIM + quad * 4) = cvec;
        v4f d = xvec - cvec;
        local += d[0]*d[0] + d[1]*d[1] + d[2]*d[2] + d[3]*d[3];
    }
    if (tid < TOK_PER_BLOCK) atomicAdd(&counts[idx_sh[tid]], 1u);

    red[tid] = local;
    __syncthreads();
#pragma unroll
    for (int s2 = 128; s2 > 0; s2 >>= 1) {
        if (tid < s2) red[tid] += red[tid + s2];
        __syncthreads();
    }
    if (tid == 0) atomicAdd(commit_sum, red[0]);
}

// ---------------------------------------------------------------------------
// Kernel 3: perplexity / usage / commit loss scalars (single 1024-thread block)
// ---------------------------------------------------------------------------
__global__ __launch_bounds__(1024) void vq_finalize_kernel(
    const unsigned* __restrict__ counts, const float* __restrict__ commit_sum,
    float* __restrict__ out_scalars, int K, float inv_nc) {
    __shared__ float sA[1024];
    __shared__ float s_total;
    const int t = threadIdx.x;
    const float c = (t < K) ? (float)counts[t] : 0.0f;

    // total = max(sum(counts), EPS_TOTAL)
    sA[t] = c;
    __syncthreads();
#pragma unroll
    for (int s = 512; s > 0; s >>= 1) {
        if (t < s) sA[t] += sA[t + s];
        __syncthreads();
    }
    if (t == 0) s_total = fmaxf(sA[0], 1e-6f);
    __syncthreads();
    const float total = s_total;

    // entropy sum of prob * log(prob + EPS_LOG)
    const float prob = c / total;
    sA[t] = prob * logf(prob + 1e-7f);
    __syncthreads();
#pragma unroll
    for (int s = 512; s > 0; s >>= 1) {
        if (t < s) sA[t] += sA[t + s];
        __syncthreads();
    }
    const float ppl = expf(-sA[0]);
    __syncthreads();

    // usage = mean(counts >= 1)
    sA[t] = (t < K && c >= 1.0f) ? 1.0f : 0.0f;
    __syncthreads();
#pragma unroll
    for (int s = 512; s > 0; s >>= 1) {
        if (t < s) sA[t] += sA[t + s];
        __syncthreads();
    }
    if (t == 0) {
        out_scalars[0] = (*commit_sum) * inv_nc;   // commit_loss
        out_scalars[1] = ppl;                      // perplexity
        out_scalars[2] = sA[0] / (float)K;         // usage
    }
}

// ---------------------------------------------------------------------------
extern "C" void kernel_launch(void* const* d_in, const int* in_sizes, int n_in,
                              void* d_out, int out_size, void* d_ws, size_t ws_size,
                              hipStream_t stream) {
    const float* z  = (const float*)d_in[0];   // [16,4096,64]
    const float* cb = (const float*)d_in[1];   // [1024,64]
    float* out = (float*)d_out;

    const int N = in_sizes[0] / C_DIM;   // 65536 tokens
    const int K = in_sizes[1] / C_DIM;   // 1024 codewords

    // Workspace layout
    char* ws = (char*)d_ws;
    unsigned* counts     = (unsigned*)ws;                        // K uints
    float*    ccn        = (float*)(ws + (size_t)K * 4);         // K floats
    float*    commit_sum = (float*)(ws + (size_t)K * 8);         // 1 float

    float* out_scalars = out + (size_t)N * C_DIM;

    vq_init_kernel<<<(K + 255) / 256, 256, 0, stream>>>(counts, commit_sum, K);
    vq_ccnorm_kernel<<<(K + 255) / 256, 256, 0, stream>>>(cb, ccn, K);
    vq_main_kernel<<<N / TOK_PER_BLOCK, 256, 0, stream>>>(
        z, cb, ccn, out, commit_sum, counts, K);
    vq_finalize_kernel<<<1, 1024, 0, stream>>>(
        counts, commit_sum, out_scalars, K, 1.0f / ((float)N * (float)C_DIM));
}